// GAT_64665027609093
// MI455X (gfx1250) — compile-verified
//
#include <hip/hip_runtime.h>
#include <float.h>

typedef __attribute__((ext_vector_type(2))) float v2f;
typedef __attribute__((ext_vector_type(8))) float v8f;

#define GEMM_WAVES 4  // 4 wave32 waves / block, one 16x16 n-tile each (64 cols)

// ---------------------------------------------------------------------------
// C[M x N] = A[M x K] * B[K x N], fp32, V_WMMA_F32_16X16X4_F32.
// K, N compile-time; N multiple of 64; M multiple of 16.
// grid.x = M/16, grid.y = N/64. A tile (16xK) and B panel (Kx64) staged in
// LDS with coalesced loads; inner loop is pure ds_load + wmma, no divergence.
// ---------------------------------------------------------------------------
template <int K, int N>
__global__ void gat_gemm_wmma_f32(const float* __restrict__ A,
                                  const float* __restrict__ B,
                                  float* __restrict__ C) {
  extern __shared__ float smem[];
  float* As = smem;             // [16][K]
  float* Bs = smem + 16 * K;    // 4 tiles of [K][16]
  const int mtile = blockIdx.x;
  const int colBase = blockIdx.y * 64;
  const int tid = threadIdx.x;

  // stage A tile: 16 rows x K, coalesced (K is power of two)
  for (int i = tid; i < 16 * K; i += GEMM_WAVES * 32) {
    int r = i / K, k = i % K;
    As[i] = A[(size_t)(mtile * 16 + r) * K + k];
  }
  // stage B panel: K rows x 64 cols -> 4 [K][16] tiles
  for (int i = tid; i < 64 * K; i += GEMM_WAVES * 32) {
    int k = i >> 6, c64 = i & 63;
    int t = c64 >> 4, c = c64 & 15;
    Bs[t * (K * 16) + k * 16 + c] = B[(size_t)k * N + colBase + c64];
  }
  __syncthreads();

  const int wave = tid >> 5;
  const int lane = tid & 31;
  const int m = lane & 15;
  const int kb = (lane >> 4) * 2;  // ISA f32 A/B layout: K offset 0 or 2
  const float* Aw = As + m * K;
  const float* Bw = Bs + wave * (K * 16) + (lane & 15);

  v8f acc = {};
#pragma unroll
  for (int k0 = 0; k0 < K; k0 += 4) {
    v2f a, b;
    a.x = Aw[k0 + kb];
    a.y = Aw[k0 + kb + 1];
    b.x = Bw[(k0 + kb) * 16];
    b.y = Bw[(k0 + kb + 1) * 16];
    acc = __builtin_amdgcn_wmma_f32_16x16x4_f32(false, a, false, b,
                                                (short)0, acc, false, false);
  }

  const int n = colBase + wave * 16 + (lane & 15);
  const int mrow0 = mtile * 16 + ((lane >> 4) << 3);  // VGPR r -> M=r or r+8
#pragma unroll
  for (int r = 0; r < 8; ++r)
    C[(size_t)(mrow0 + r) * N + n] = acc[r];
}

// ---------------------------------------------------------------------------
__global__ void gat_fill(float* __restrict__ p, float v, long long n) {
  long long t = (long long)blockIdx.x * blockDim.x + threadIdx.x;
  if (t < n) p[t] = v;
}

// dst[rows x dcols] = src[rows x scols] zero-padded on the right
__global__ void gat_pad_copy(const float* __restrict__ src, float* __restrict__ dst,
                             int rows, int scols, int dcols) {
  int t = blockIdx.x * blockDim.x + threadIdx.x;
  if (t >= rows * dcols) return;
  int r = t / dcols, c = t - r * dcols;
  dst[t] = (c < scols) ? src[r * scols + c] : 0.f;
}

// per-(node,head) attention logits: a = sum_c h[n, hh*Cc + c] * att[hh, c]
__global__ void gat_node_logits(const float* __restrict__ h, int ldh,
                                const float* __restrict__ att_src,
                                const float* __restrict__ att_dst,
                                float* __restrict__ asrc, float* __restrict__ adst,
                                int n, int H, int Cc) {
  int i = blockIdx.x * blockDim.x + threadIdx.x;
  if (i >= n * H) return;
  int node = i / H, hh = i - node * H;
  const float* hp = h + (size_t)node * ldh + hh * Cc;
  float s1 = 0.f, s2 = 0.f;
  for (int c = 0; c < Cc; ++c) {
    float v = hp[c];
    s1 += v * att_src[hh * Cc + c];
    s2 += v * att_dst[hh * Cc + c];
  }
  asrc[i] = s1;
  adst[i] = s2;
}

// pass A: e = leaky_relu(asrc[src]+adst[dst]); segment max over dst
template <int H>
__global__ void gat_edge_logits(const int* __restrict__ src,
                                const int* __restrict__ dst,
                                int E_, int n,
                                const float* __restrict__ asrc,
                                const float* __restrict__ adst,
                                float* __restrict__ elog,
                                float* __restrict__ mmax) {
  long long t = (long long)blockIdx.x * blockDim.x + threadIdx.x;
  long long tot = (long long)(E_ + n) * H;
  if (t >= tot) return;
  int e = (int)(t / H), hh = (int)(t - (long long)e * H);
  int s, d;
  if (e < E_) { s = src[e]; d = dst[e]; } else { s = d = e - E_; }
  float v = asrc[s * H + hh] + adst[d * H + hh];
  v = (v > 0.f) ? v : 0.2f * v;
  elog[t] = v;
  // native float atomic max (global_atomic_max_num_f32)
  __hip_atomic_fetch_max(&mmax[d * H + hh], v, __ATOMIC_RELAXED,
                         __HIP_MEMORY_SCOPE_AGENT);
}

// pass B: w = exp(e - m[dst]); segment sum over dst
template <int H>
__global__ void gat_edge_exp(const int* __restrict__ dst,
                             int E_, int n,
                             float* __restrict__ elog,
                             const float* __restrict__ mmax,
                             float* __restrict__ ssum) {
  long long t = (long long)blockIdx.x * blockDim.x + threadIdx.x;
  long long tot = (long long)(E_ + n) * H;
  if (t >= tot) return;
  int e = (int)(t / H), hh = (int)(t - (long long)e * H);
  int d = (e < E_) ? dst[e] : (e - E_);
  float w = __expf(elog[t] - mmax[d * H + hh]);
  elog[t] = w;
  atomicAdd(&ssum[d * H + hh], w);  // native global_atomic_add_f32
}

// pass C: acc[dst, hh, c] += h[src, hh, c] * alpha.  One block per edge.
template <int H, int Cc>
__global__ void gat_edge_scatter(const int* __restrict__ src,
                                 const int* __restrict__ dst,
                                 int E_, int n,
                                 const float* __restrict__ elog,
                                 const float* __restrict__ ssum,
                                 const float* __restrict__ hfeat, int ldh,
                                 float* __restrict__ acc) {
  const int F = H * Cc;
  int e = blockIdx.x;
  int f = threadIdx.x;
  if (f >= F) return;
  int s, d;
  if (e < E_) { s = src[e]; d = dst[e]; } else { s = d = e - E_; }
  const float* gp = hfeat + (size_t)s * ldh + f;
  __builtin_prefetch(gp, 0, 0);  // global_prefetch_b8: head start on gather
  int hh = f / Cc;
  float alpha = elog[(long long)e * H + hh] / (ssum[d * H + hh] + 1e-16f);
  atomicAdd(&acc[(size_t)d * F + f], *gp * alpha);
}

// acc = elu(acc + bias), in place
__global__ void gat_bias_elu(float* __restrict__ acc,
                             const float* __restrict__ bias, long long n, int F) {
  long long t = (long long)blockIdx.x * blockDim.x + threadIdx.x;
  if (t >= n) return;
  float v = acc[t] + bias[(int)(t % F)];
  acc[t] = (v > 0.f) ? v : (__expf(v) - 1.f);
}

// out = log_softmax(acc2 + b2) per node (NCLASS = 40)
__global__ void gat_final_logsoftmax(const float* __restrict__ acc2,
                                     const float* __restrict__ b2,
                                     float* __restrict__ out, int n, int Ccls) {
  int i = blockIdx.x * blockDim.x + threadIdx.x;
  if (i >= n) return;
  float v[64];
  float mx = -FLT_MAX;
  for (int c = 0; c < Ccls; ++c) {
    float x = acc2[(size_t)i * Ccls + c] + b2[c];
    v[c] = x;
    mx = (x > mx) ? x : mx;
  }
  float s = 0.f;
  for (int c = 0; c < Ccls; ++c) s += __expf(v[c] - mx);
  float lse = logf(s);
  for (int c = 0; c < Ccls; ++c) out[(size_t)i * Ccls + c] = v[c] - mx - lse;
}

// ---------------------------------------------------------------------------
extern "C" void kernel_launch(void* const* d_in, const int* in_sizes, int n_in,
                              void* d_out, int out_size, void* d_ws, size_t ws_size,
                              hipStream_t stream) {
  const float* x   = (const float*)d_in[0];
  const int*   ei  = (const int*)  d_in[1];
  const float* W1  = (const float*)d_in[2];
  const float* a1s = (const float*)d_in[3];
  const float* a1d = (const float*)d_in[4];
  const float* b1  = (const float*)d_in[5];
  const float* W2  = (const float*)d_in[6];
  const float* a2s = (const float*)d_in[7];
  const float* a2d = (const float*)d_in[8];
  const float* b2  = (const float*)d_in[9];
  float* out = (float*)d_out;

  const int N_ = 50000, E_ = 800000;
  const int K1 = 128, H1 = 8, C1 = 32, F1 = 256;   // layer 1
  const int K2 = 256, C2 = 40, F2 = 40, P2 = 64;   // layer 2 (padded to 64)
  const int EN = E_ + N_;
  const int* src = ei;
  const int* dst = ei + E_;

  float* ws = (float*)d_ws;
  size_t off = 0;
  auto alloc = [&](size_t nf) { float* p = ws + off; off += nf; return p; };
  float* h1    = alloc((size_t)N_ * F1);
  float* asrc1 = alloc((size_t)N_ * H1);
  float* adst1 = alloc((size_t)N_ * H1);
  float* m1    = alloc((size_t)N_ * H1);
  float* s1    = alloc((size_t)N_ * H1);
  float* e1    = alloc((size_t)EN * H1);
  float* acc1  = alloc((size_t)N_ * F1);   // elu output = layer-2 input
  float* W2p   = alloc((size_t)K2 * P2);   // W2 zero-padded 40 -> 64 cols
  float* h2p   = alloc((size_t)N_ * P2);   // padded layer-2 features
  float* asrc2 = alloc((size_t)N_);
  float* adst2 = alloc((size_t)N_);
  float* m2    = alloc((size_t)N_);
  float* s2    = alloc((size_t)N_);
  float* e2    = alloc((size_t)EN);
  float* acc2  = alloc((size_t)N_ * F2);

  auto blocks = [](long long n, int b) { return (unsigned)((n + b - 1) / b); };

  // init accumulators / segment state (every call: graph-replay safe)
  gat_fill<<<blocks((long long)N_ * H1, 256), 256, 0, stream>>>(m1, -FLT_MAX, (long long)N_ * H1);
  gat_fill<<<blocks((long long)N_ * H1, 256), 256, 0, stream>>>(s1, 0.f, (long long)N_ * H1);
  gat_fill<<<blocks((long long)N_ * F1, 256), 256, 0, stream>>>(acc1, 0.f, (long long)N_ * F1);
  gat_fill<<<blocks(N_, 256), 256, 0, stream>>>(m2, -FLT_MAX, N_);
  gat_fill<<<blocks(N_, 256), 256, 0, stream>>>(s2, 0.f, N_);
  gat_fill<<<blocks((long long)N_ * F2, 256), 256, 0, stream>>>(acc2, 0.f, (long long)N_ * F2);
  gat_pad_copy<<<blocks((long long)K2 * P2, 256), 256, 0, stream>>>(W2, W2p, K2, C2, P2);

  // ----- layer 1: h1 = x @ W1 -----
  gat_gemm_wmma_f32<K1, F1><<<dim3(N_ / 16, F1 / 64), GEMM_WAVES * 32,
                              80 * K1 * sizeof(float), stream>>>(x, W1, h1);
  gat_node_logits<<<blocks((long long)N_ * H1, 256), 256, 0, stream>>>(
      h1, F1, a1s, a1d, asrc1, adst1, N_, H1, C1);
  gat_edge_logits<H1><<<blocks((long long)EN * H1, 256), 256, 0, stream>>>(
      src, dst, E_, N_, asrc1, adst1, e1, m1);
  gat_edge_exp<H1><<<blocks((long long)EN * H1, 256), 256, 0, stream>>>(
      dst, E_, N_, e1, m1, s1);
  gat_edge_scatter<H1, C1><<<EN, F1, 0, stream>>>(
      src, dst, E_, N_, e1, s1, h1, F1, acc1);
  gat_bias_elu<<<blocks((long long)N_ * F1, 256), 256, 0, stream>>>(
      acc1, b1, (long long)N_ * F1, F1);

  // ----- layer 2: h2p = acc1 @ W2p (padded, guard-free) -----
  gat_gemm_wmma_f32<K2, P2><<<dim3(N_ / 16, 1), GEMM_WAVES * 32,
                              80 * K2 * sizeof(float), stream>>>(acc1, W2p, h2p);
  gat_node_logits<<<blocks(N_, 256), 256, 0, stream>>>(
      h2p, P2, a2s, a2d, asrc2, adst2, N_, 1, C2);
  gat_edge_logits<1><<<blocks(EN, 256), 256, 0, stream>>>(
      src, dst, E_, N_, asrc2, adst2, e2, m2);
  gat_edge_exp<1><<<blocks(EN, 256), 256, 0, stream>>>(
      dst, E_, N_, e2, m2, s2);
  gat_edge_scatter<1, C2><<<EN, 64, 0, stream>>>(
      src, dst, E_, N_, e2, s2, h2p, P2, acc2);

  gat_final_logsoftmax<<<blocks(N_, 128), 128, 0, stream>>>(acc2, b2, out, N_, F2);
}